// SPC_85469849190654
// MI455X (gfx1250) — compile-verified
//
#include <hip/hip_runtime.h>

#define AS1 __attribute__((address_space(1)))
#define AS3 __attribute__((address_space(3)))

typedef int v4i __attribute__((vector_size(16)));

// --- CDNA5 async Global->LDS gather (ASYNCcnt path) -------------------------
// Probe-confirmed: builtin exists; param0 is 'v4i AS1*' (global src), so the
// order is (global src, lds dst, imm offset, imm cpol).
#if __has_builtin(__builtin_amdgcn_global_load_async_to_lds_b128)
#define ASYNC_GATHER_B128(g, l)                                                \
  __builtin_amdgcn_global_load_async_to_lds_b128((AS1 v4i*)(g), (AS3 v4i*)(l), \
                                                 0, 0)
#else
#define ASYNC_GATHER_B128(g, l)                                                \
  asm volatile("global_load_async_to_lds_b128 %0, %1, off" ::                  \
                   "v"((unsigned)(size_t)(AS3 v4i*)(l)),                       \
                   "v"((const void*)(g))                                       \
               : "memory")
#endif

#if __has_builtin(__builtin_amdgcn_s_wait_asynccnt)
#define WAIT_ASYNC(n) __builtin_amdgcn_s_wait_asynccnt(n)
#else
#define WAIT_ASYNC(n) asm volatile("s_wait_asynccnt %0" ::"i"(n) : "memory")
#endif

// One wave per point: lane = feature (F==32 == wave32 width).
// Per point the gather is 8 rows x 128 B = 64 chunks of 16 B = 2 async B128
// wave-instructions. Double-buffered in LDS per wave (2 x 1 KB).
__global__ __launch_bounds__(256) void spc_trilinear_gather(
    const float* __restrict__ coeffs,  // [N,8]
    const int* __restrict__ cidx,      // [N,8]
    const float* __restrict__ feats,   // [V,32]
    float* __restrict__ out,           // [N,32]
    int N) {
  __shared__ float smem[8 * 2 * 256];  // 8 waves * 2 buffers * 256 floats = 16 KB
  const int lane = threadIdx.x & 31;
  const int wave = __builtin_amdgcn_readfirstlane((int)(threadIdx.x >> 5));
  AS3 float* wbuf = ((AS3 float*)smem) + wave * 512;

  const int totalWaves = (int)((gridDim.x * blockDim.x) >> 5);
  const int sub = lane >> 3;  // which row within the 4-row half (0..3)
  const int q = lane & 7;     // 16-byte chunk within a 128 B row (0..7)

  int n = blockIdx.x * 8 + wave;  // wave-uniform point id (scalar)
  if (n < N) {
    // Prologue: issue async gather for first point into buffer 0.
    int r0 = cidx[n * 8 + sub];      // rows 0..3 (lanes 0..31 -> chunks 0..31)
    int r1 = cidx[n * 8 + 4 + sub];  // rows 4..7 (chunks 32..63)
    ASYNC_GATHER_B128(feats + (size_t)r0 * 32 + q * 4, wbuf + lane * 4);
    ASYNC_GATHER_B128(feats + (size_t)r1 * 32 + q * 4, wbuf + 128 + lane * 4);
  }

  int buf = 0;
  for (; n < N; n += totalWaves) {
    const int nn = n + totalWaves;
    if (nn < N) {
      // Issue next point's gather into the other buffer, then wait until the
      // two oldest async ops (current point) have landed in LDS.
      const int ob = buf ^ 1;
      int r0 = cidx[nn * 8 + sub];
      int r1 = cidx[nn * 8 + 4 + sub];
      ASYNC_GATHER_B128(feats + (size_t)r0 * 32 + q * 4,
                        wbuf + ob * 256 + lane * 4);
      ASYNC_GATHER_B128(feats + (size_t)r1 * 32 + q * 4,
                        wbuf + ob * 256 + 128 + lane * 4);
      WAIT_ASYNC(2);
    } else {
      WAIT_ASYNC(0);
    }

    // Consume: lane = feature, 8 conflict-free ds_load_b32 + FMA.
    // coeffs[n*8+k] has a wave-uniform address -> scalar (SMEM) loads.
    float acc = 0.0f;
    const AS3 float* src = wbuf + buf * 256;
#pragma unroll
    for (int k = 0; k < 8; ++k) {
      float c = coeffs[n * 8 + k];
      c = fminf(fmaxf(c, 0.0f), 1.0f);
      acc = fmaf(c, src[k * 32 + lane], acc);
    }
    // Output is write-once; keep the 192 MB L2 reserved for the 128 MB
    // feature table -> non-temporal store.
    __builtin_nontemporal_store(acc, out + (size_t)n * 32 + lane);
    buf ^= 1;
  }
}

extern "C" void kernel_launch(void* const* d_in, const int* in_sizes, int n_in,
                              void* d_out, int out_size, void* d_ws,
                              size_t ws_size, hipStream_t stream) {
  const float* coeffs = (const float*)d_in[0];
  const int* cidx = (const int*)d_in[1];
  const float* feats = (const float*)d_in[2];
  float* out = (float*)d_out;

  const int N = in_sizes[0] / 8;  // coeffs is [N,8]

  // Enough waves to fill the chip, few enough that each wave loops several
  // times so the async double-buffer pipeline actually overlaps.
  int blocks = (N + 7) / 8;
  if (blocks > 4096) blocks = 4096;
  if (blocks < 1) blocks = 1;

  hipLaunchKernelGGL(spc_trilinear_gather, dim3(blocks), dim3(256), 0, stream,
                     coeffs, cidx, feats, out, N);
}